// Transformer_attn_25383256719651
// MI455X (gfx1250) — compile-verified
//
#include <hip/hip_runtime.h>
#include <math.h>

typedef __attribute__((ext_vector_type(2))) float v2f;
typedef __attribute__((ext_vector_type(8))) float v8f;

#define CCH   96
#define HGT   128
#define WID   128
#define NB    32
#define LDS_STRIDE 66   // 64 pixels + 2 pad (keeps b32 pairs 8B-aligned, spreads banks)

// ---------------- Kernel 0: M = convq^T @ convk (and its transpose) ----------------
// scores use x_n^T (Q^T K) x_m, so precompute M[c,c'] = sum_o Q[o,c] K[o,c'].
__global__ void prep_M(const float* __restrict__ Q, const float* __restrict__ K,
                       float* __restrict__ M, float* __restrict__ Mt) {
    int idx = blockIdx.x * blockDim.x + threadIdx.x;
    if (idx >= CCH * CCH) return;
    int c = idx / CCH, cc = idx % CCH;
    float s = 0.f;
    for (int o = 0; o < CCH; ++o) s += Q[o * CCH + c] * K[o * CCH + cc];
    M[c * CCH + cc]  = s;
    Mt[cc * CCH + c] = s;
}

__global__ void zero_scores(float* s) { s[threadIdx.x] = 0.f; }

// ---------------- Kernel 1: Gram-based scores via V_WMMA_F32_16X16X4_F32 ----------------
// grid = NB * 2(parity) * 32(chunks of 64 pixels); block = 192 threads = 6 waves.
// Wave i owns row-tile i (channels 16i..16i+15) of Grams C_aa, C_ab, C_bb (96x96, K=2048).
// scores layout: scores[b][n][0] = s(n,n), scores[b][n][1] = s(n, n^2)
__global__ void __launch_bounds__(192)
score_gram(const float* __restrict__ x, const float* __restrict__ M,
           const float* __restrict__ Mt, float* __restrict__ scores) {
    int blk   = blockIdx.x;
    int chunk = blk & 31;        // 64-pixel chunk: hh in {2*chunk, 2*chunk+1}, all 32 j
    int pi    = (blk >> 5) & 1;  // parity class: patches {pi, pi+2}
    int b     = blk >> 6;

    __shared__ float smem[2 * CCH * LDS_STRIDE];

    int tid = threadIdx.x;
    // Stage: 2 patches x 96 channels x 64 pixels. Valid columns: w = pi*65 + 2*j.
    const int TOT = 2 * CCH * 64;
    for (int e = tid; e < TOT; e += 192) {
        int q     = e & 63;
        int c     = (e >> 6) % CCH;
        int patch = e / (64 * CCH);
        int hh    = chunk * 2 + (q >> 5);
        int j     = q & 31;
        int h     = patch * 64 + hh;
        int w     = pi * 65 + 2 * j;
        smem[(patch * CCH + c) * LDS_STRIDE + q] =
            x[(((size_t)b * CCH + c) * HGT + h) * WID + w];
    }
    __syncthreads();

    int wave = tid >> 5;     // 0..5 : Gram row-tile
    int lane = tid & 31;
    int lrow = lane & 15;
    int lhi  = lane >> 4;

    v8f acc_aa[6], acc_ab[6], acc_bb[6];
#pragma unroll
    for (int j6 = 0; j6 < 6; ++j6) {
        acc_aa[j6] = (v8f){0.f,0.f,0.f,0.f,0.f,0.f,0.f,0.f};
        acc_ab[j6] = (v8f){0.f,0.f,0.f,0.f,0.f,0.f,0.f,0.f};
        acc_bb[j6] = (v8f){0.f,0.f,0.f,0.f,0.f,0.f,0.f,0.f};
    }

    // 16 K-steps of 4 pixels. For 32-bit operands the A-operand layout
    // (m,k)->(lane=m+16*(k>=2), vgpr=k&1) equals the B-operand layout with m<->n,
    // so one v2f per channel-tile serves as both A and B.
    for (int kp = 0; kp < 16; ++kp) {
        int qb = kp * 4 + 2 * lhi;   // this half-wave covers pixels qb, qb+1
        v2f TA[6], TB[6];
#pragma unroll
        for (int j6 = 0; j6 < 6; ++j6) {
            int ca = 16 * j6 + lrow;
            const float* pa = &smem[(0 * CCH + ca) * LDS_STRIDE + qb];
            const float* pb = &smem[(1 * CCH + ca) * LDS_STRIDE + qb];
            TA[j6].x = pa[0]; TA[j6].y = pa[1];
            TB[j6].x = pb[0]; TB[j6].y = pb[1];
        }
        // this wave's A-operands (row tile = wave); direct loads keep arrays in regs
        int ci = 16 * wave + lrow;
        v2f Ai, Bi;
        Ai.x = smem[(0 * CCH + ci) * LDS_STRIDE + qb];
        Ai.y = smem[(0 * CCH + ci) * LDS_STRIDE + qb + 1];
        Bi.x = smem[(1 * CCH + ci) * LDS_STRIDE + qb];
        Bi.y = smem[(1 * CCH + ci) * LDS_STRIDE + qb + 1];

#pragma unroll
        for (int j6 = 0; j6 < 6; ++j6) {
            acc_aa[j6] = __builtin_amdgcn_wmma_f32_16x16x4_f32(
                false, Ai, false, TA[j6], (short)0, acc_aa[j6], false, false);
            acc_ab[j6] = __builtin_amdgcn_wmma_f32_16x16x4_f32(
                false, Ai, false, TB[j6], (short)0, acc_ab[j6], false, false);
            acc_bb[j6] = __builtin_amdgcn_wmma_f32_16x16x4_f32(
                false, Bi, false, TB[j6], (short)0, acc_bb[j6], false, false);
        }
    }

    // Fold Gram stripe against M (and M^T for the transposed cross term).
    float paa = 0.f, pab = 0.f, pba = 0.f, pbb = 0.f;
#pragma unroll
    for (int j6 = 0; j6 < 6; ++j6) {
#pragma unroll
        for (int g = 0; g < 8; ++g) {
            int row = 16 * wave + g + 8 * lhi;  // C/D layout: vgpr g -> M=g (+8 for hi half)
            int col = 16 * j6 + lrow;
            float m  = M[row * CCH + col];
            float mt = Mt[row * CCH + col];
            paa += m  * acc_aa[j6][g];
            pab += m  * acc_ab[j6][g];
            pba += mt * acc_ab[j6][g];
            pbb += m  * acc_bb[j6][g];
        }
    }
    for (int off = 16; off; off >>= 1) {
        paa += __shfl_xor(paa, off, 32);
        pab += __shfl_xor(pab, off, 32);
        pba += __shfl_xor(pba, off, 32);
        pbb += __shfl_xor(pbb, off, 32);
    }
    if (lane == 0) {
        atomicAdd(&scores[(b * 4 + pi) * 2 + 0], paa);       // s(pi,pi)
        atomicAdd(&scores[(b * 4 + pi) * 2 + 1], pab);       // s(pi,pi+2)
        atomicAdd(&scores[(b * 4 + pi + 2) * 2 + 0], pbb);   // s(pi+2,pi+2)
        atomicAdd(&scores[(b * 4 + pi + 2) * 2 + 1], pba);   // s(pi+2,pi)
    }
}

// ---------------- Kernel 2: output = mask ? x : a_nn*x + a_nn'*x_partner ----------------
// Each thread handles a float4 at h<64 AND its partner row h+64 (halves the reads).
__global__ void __launch_bounds__(256)
apply_out(const float* __restrict__ x, const float* __restrict__ scores,
          const float* __restrict__ o1p, const float* __restrict__ o2p,
          const float* __restrict__ o3p, float* __restrict__ out) {
    size_t t = (size_t)blockIdx.x * blockDim.x + threadIdx.x;
    const size_t NT = (size_t)NB * CCH * 64 * (WID / 4);
    if (t >= NT) return;
    int w4 = (int)(t % 32);
    int h  = (int)((t / 32) % 64);
    int c  = (int)((t / (32 * 64)) % CCH);
    int b  = (int)(t / (32 * 64 * CCH));
    int wblock = (w4 >= 16) ? 1 : 0;
    int ntop = wblock, nbot = 2 + wblock;

    float o1 = o1p[0], o2 = o2p[0], o3 = o3p[0];
    // attn(n,n) = lrelu(s+o2)+o3+o1 ; attn(n,n^2) = lrelu(s+o2)+o3
    float sd, so;
    sd = scores[(b * 4 + ntop) * 2 + 0] + o2; sd = sd >= 0.f ? sd : 1e-4f * sd;
    so = scores[(b * 4 + ntop) * 2 + 1] + o2; so = so >= 0.f ? so : 1e-4f * so;
    float dt = sd + o3 + o1, ot = so + o3;
    sd = scores[(b * 4 + nbot) * 2 + 0] + o2; sd = sd >= 0.f ? sd : 1e-4f * sd;
    so = scores[(b * 4 + nbot) * 2 + 1] + o2; so = so >= 0.f ? so : 1e-4f * so;
    float db = sd + o3 + o1, ob = so + o3;

    size_t base_t = (((size_t)b * CCH + c) * HGT + h) * WID + (size_t)w4 * 4;
    size_t base_b = base_t + (size_t)64 * WID;
    float4 xt = *(const float4*)(x + base_t);
    float4 xb = *(const float4*)(x + base_b);
    float at[4] = {xt.x, xt.y, xt.z, xt.w};
    float ab[4] = {xb.x, xb.y, xb.z, xb.w};
    float rt[4], rb[4];
#pragma unroll
    for (int e = 0; e < 4; ++e) {
        int w = w4 * 4 + e;
        bool mask1 = ((wblock + w) & 1) == 0;  // checkerboard collapses to f(w)
        rt[e] = mask1 ? at[e] : dt * at[e] + ot * ab[e];
        rb[e] = mask1 ? ab[e] : db * ab[e] + ob * at[e];
    }
    *(float4*)(out + base_t) = make_float4(rt[0], rt[1], rt[2], rt[3]);
    *(float4*)(out + base_b) = make_float4(rb[0], rb[1], rb[2], rb[3]);
}

// ---------------- Kernel 3: logdet of the two 2x2 parity blocks ----------------
__global__ void logdet_k(const float* __restrict__ scores,
                         const float* __restrict__ o1p, const float* __restrict__ o2p,
                         const float* __restrict__ o3p, float* __restrict__ out_ld) {
    int b = threadIdx.x;
    if (b >= NB) return;
    float o1 = o1p[0], o2 = o2p[0], o3 = o3p[0];
    float a[4][2];
#pragma unroll
    for (int n = 0; n < 4; ++n) {
#pragma unroll
        for (int k = 0; k < 2; ++k) {
            float s = scores[(b * 4 + n) * 2 + k] + o2;
            s = s >= 0.f ? s : 1e-4f * s;
            a[n][k] = s + o3 + (k == 0 ? o1 : 0.f);
        }
    }
    float d1 = a[0][0] * a[2][0] - a[0][1] * a[2][1];  // [[a00,a02],[a20,a22]]
    float d2 = a[1][0] * a[3][0] - a[1][1] * a[3][1];  // [[a11,a13],[a31,a33]]
    out_ld[b] = (logf(fabsf(d1)) + logf(fabsf(d2))) * 196608.0f;  // p*(p/2)*C
}

extern "C" void kernel_launch(void* const* d_in, const int* in_sizes, int n_in,
                              void* d_out, int out_size, void* d_ws, size_t ws_size,
                              hipStream_t stream) {
    (void)in_sizes; (void)n_in; (void)out_size; (void)ws_size;
    const float* x  = (const float*)d_in[0];
    const float* cq = (const float*)d_in[1];
    const float* ck = (const float*)d_in[2];
    const float* o1 = (const float*)d_in[3];
    const float* o2 = (const float*)d_in[4];
    const float* o3 = (const float*)d_in[5];
    float* out = (float*)d_out;

    float* M      = (float*)d_ws;          // 96*96
    float* Mt     = M + CCH * CCH;         // 96*96
    float* scores = Mt + CCH * CCH;        // 32*8

    prep_M<<<(CCH * CCH + 255) / 256, 256, 0, stream>>>(cq, ck, M, Mt);
    zero_scores<<<1, NB * 8, 0, stream>>>(scores);
    score_gram<<<NB * 2 * 32, 192, 0, stream>>>(x, M, Mt, scores);

    const size_t NT = (size_t)NB * CCH * 64 * (WID / 4);
    apply_out<<<(unsigned)((NT + 255) / 256), 256, 0, stream>>>(x, scores, o1, o2, o3, out);
    logdet_k<<<1, 32, 0, stream>>>(scores, o1, o2, o3,
                                   out + (size_t)NB * CCH * HGT * WID);
}